// Block_43911745634528
// MI455X (gfx1250) — compile-verified
//
#include <hip/hip_runtime.h>
#include <hip/hip_bf16.h>
#include <math.h>

// ---------------------------------------------------------------------------
// Transformer block (pre-LN attn + FFN) for gfx1250 using fp32 WMMA
// (V_WMMA_F32_16X16X4_F32). One wave (32 lanes) owns each output tile.
// GEMMs use 32x64 tiles (two A fragments per B fragment) to double the
// FLOP/byte intensity out of L2 (the binding resource: weights live in the
// 192MB L2, and B-fragment traffic dominates at 16-row tiles).
// ---------------------------------------------------------------------------

#define BB  2
#define TT  2048
#define CC  512
#define HH  16
#define HSZ 32
#define DFF 2048
#define LN_EPS 1e-3f

typedef float v2f __attribute__((ext_vector_type(2)));
typedef float v8f __attribute__((ext_vector_type(8)));

// D = A(16x4 f32) * B(4x16 f32) + C(16x16 f32)
__device__ __forceinline__ v8f wmma_f32(v2f a, v2f b, v8f c) {
    return __builtin_amdgcn_wmma_f32_16x16x4_f32(
        /*neg_a=*/false, a, /*neg_b=*/false, b,
        /*c_mod=*/(short)0, c, /*reuse_a=*/false, /*reuse_b=*/false);
}

// A-fragment: 16x4 tile from row-major src (rows at `base`, leading dim ld).
// lane<16 -> (row=l16, K=k0..k0+1), lane>=16 -> (row=l16, K=k0+2..k0+3)
__device__ __forceinline__ v2f load_a_frag(const float* base, int ld, int k0,
                                           int l16, int hiOff) {
    const float* p = base + (size_t)l16 * ld + k0 + hiOff;
    v2f a; a.x = p[0]; a.y = p[1];
    return a;
}

// B-fragment: 4x16 tile of row-major W (ld = row stride), cols [col0,col0+16)
// lanes index N; halves index K (lane<16 -> K=k0/k0+1, lane>=16 -> K=k0+2/+3)
__device__ __forceinline__ v2f load_b_frag(const float* W, int ld, int k0,
                                           int col0, int l16, int hiOff) {
    const float* p = W + (size_t)(k0 + hiOff) * ld + col0 + l16;
    v2f b; b.x = p[0]; b.y = p[ld];
    return b;
}

// ---------------------------------------------------------------------------
// LayerNorm: one wave per row of C=512
// ---------------------------------------------------------------------------
__global__ void ln_kernel(const float* __restrict__ x, const float* __restrict__ g,
                          const float* __restrict__ be, float* __restrict__ h) {
    const int row  = blockIdx.x;
    const int lane = threadIdx.x;
    const float4* xr = (const float4*)(x + (size_t)row * CC);
    float4 vals[4];
    float s = 0.f, s2 = 0.f;
#pragma unroll
    for (int i = 0; i < 4; ++i) {
        float4 t = xr[lane + i * 32];
        vals[i] = t;
        s  += t.x + t.y + t.z + t.w;
        s2 += t.x * t.x + t.y * t.y + t.z * t.z + t.w * t.w;
    }
#pragma unroll
    for (int m = 1; m < 32; m <<= 1) {
        s  += __shfl_xor(s, m);
        s2 += __shfl_xor(s2, m);
    }
    const float mu   = s * (1.0f / (float)CC);
    const float var  = s2 * (1.0f / (float)CC) - mu * mu;
    const float rstd = rsqrtf(var + LN_EPS);
    const float4* gr = (const float4*)g;
    const float4* br = (const float4*)be;
    float4* hr = (float4*)(h + (size_t)row * CC);
#pragma unroll
    for (int i = 0; i < 4; ++i) {
        float4 gv = gr[lane + i * 32];
        float4 bv = br[lane + i * 32];
        float4 t  = vals[i];
        float4 r;
        r.x = (t.x - mu) * rstd * gv.x + bv.x;
        r.y = (t.y - mu) * rstd * gv.y + bv.y;
        r.z = (t.z - mu) * rstd * gv.z + bv.z;
        r.w = (t.w - mu) * rstd * gv.w + bv.w;
        hr[lane + i * 32] = r;
    }
}

// ---------------------------------------------------------------------------
// QKV projection: out[b,h,t,d] = sum_c h[b,t,c]*W[h,c,d] + bias[h,d]
// grid = (BT/32, H, 3); one wave computes a 32x32 tile for one head
// (two 16-row sub-tiles sharing every B fragment).
// ---------------------------------------------------------------------------
__global__ void qkv_kernel(const float* __restrict__ hin,
                           const float* __restrict__ Wq, const float* __restrict__ Wk,
                           const float* __restrict__ Wv,
                           const float* __restrict__ bq, const float* __restrict__ bk,
                           const float* __restrict__ bv,
                           float* __restrict__ qo, float* __restrict__ ko,
                           float* __restrict__ vo) {
    const float* W; const float* bias; float* out;
    if (blockIdx.z == 0)      { W = Wq; bias = bq; out = qo; }
    else if (blockIdx.z == 1) { W = Wk; bias = bk; out = ko; }
    else                      { W = Wv; bias = bv; out = vo; }

    const int lane   = threadIdx.x;
    const int l16    = lane & 15;
    const int hiOff  = (lane >> 4) ? 2 : 0;
    const int rowOff = (lane >> 4) ? 8 : 0;
    const int head   = blockIdx.y;
    const int m0     = blockIdx.x * 32;

    const float* Wh = W + (size_t)head * CC * HSZ;   // [C, HS] row-major
    const float* A0 = hin + (size_t)m0 * CC;
    const float* A1 = A0 + (size_t)16 * CC;

    v8f c00 = {}, c01 = {}, c10 = {}, c11 = {};
    for (int k0 = 0; k0 < CC; k0 += 4) {
        v2f a0 = load_a_frag(A0, CC, k0, l16, hiOff);
        v2f a1 = load_a_frag(A1, CC, k0, l16, hiOff);
        v2f b0 = load_b_frag(Wh, HSZ, k0, 0,  l16, hiOff);
        v2f b1 = load_b_frag(Wh, HSZ, k0, 16, l16, hiOff);
        c00 = wmma_f32(a0, b0, c00);
        c01 = wmma_f32(a0, b1, c01);
        c10 = wmma_f32(a1, b0, c10);
        c11 = wmma_f32(a1, b1, c11);
    }
    const float bb0 = bias[head * HSZ + l16];
    const float bb1 = bias[head * HSZ + 16 + l16];
#pragma unroll
    for (int s = 0; s < 2; ++s) {
        const v8f& d0 = s ? c10 : c00;
        const v8f& d1 = s ? c11 : c01;
#pragma unroll
        for (int j = 0; j < 8; ++j) {
            const int m = m0 + s * 16 + j + rowOff;
            const int b = m / TT;
            const int t = m - b * TT;
            float* orow = out + (((size_t)b * HH + head) * TT + t) * HSZ;
            orow[l16]      = d0[j] + bb0;
            orow[16 + l16] = d1[j] + bb1;
        }
    }
}

// ---------------------------------------------------------------------------
// Flash-style causal attention, one wave per 16-query tile per (b,head).
// Scores never hit HBM (would be 536 MB). Online softmax state lives in the
// half-wave that owns each row of the WMMA C/D layout.
// ---------------------------------------------------------------------------
__global__ void attn_kernel(const float* __restrict__ q, const float* __restrict__ k,
                            const float* __restrict__ v, float* __restrict__ o) {
    __shared__ float lp[16 * 17];   // padded P tile (C/D layout -> A layout)
    const int lane   = threadIdx.x;
    const int l16    = lane & 15;
    const int hiOff  = (lane >> 4) ? 2 : 0;
    const int rowOff = (lane >> 4) ? 8 : 0;
    const int t0     = blockIdx.x * 16;
    const int bh     = blockIdx.y;
    const int b      = bh / HH;
    const int head   = bh - b * HH;

    const float* qb = q + ((size_t)bh * TT + t0) * HSZ;
    v2f qa[8];
#pragma unroll
    for (int kk = 0; kk < 8; ++kk)
        qa[kk] = load_a_frag(qb, HSZ, kk * 4, l16, hiOff);

    float mrow[8], lrow[8];
#pragma unroll
    for (int j = 0; j < 8; ++j) { mrow[j] = -INFINITY; lrow[j] = 0.f; }
    v8f o0 = {}, o1 = {};

    for (int s0 = 0; s0 <= t0; s0 += 16) {
        // ---- S = Q * K^T  (K^T as B-fragment == A-pattern load of K) ----
        const float* kb = k + ((size_t)bh * TT + s0) * HSZ;
        v8f sc = {};
#pragma unroll
        for (int kk = 0; kk < 8; ++kk) {
            v2f bf = load_a_frag(kb, HSZ, kk * 4, l16, hiOff);
            sc = wmma_f32(qa[kk], bf, sc);
        }
        if (s0 == t0) {   // diagonal tile: col (s0+l16) > row (t0+j+rowOff)
#pragma unroll
            for (int j = 0; j < 8; ++j)
                if (l16 > j + rowOff) sc[j] = -INFINITY;
        }
        // ---- online softmax (rows live in 16-lane halves) ----
        float p[8];
#pragma unroll
        for (int j = 0; j < 8; ++j) {
            float rmax = sc[j];
            rmax = fmaxf(rmax, __shfl_xor(rmax, 1));
            rmax = fmaxf(rmax, __shfl_xor(rmax, 2));
            rmax = fmaxf(rmax, __shfl_xor(rmax, 4));
            rmax = fmaxf(rmax, __shfl_xor(rmax, 8));
            const float mnew  = fmaxf(mrow[j], rmax);
            const float alpha = __expf(mrow[j] - mnew);
            const float pj    = __expf(sc[j] - mnew);
            float rsum = pj;
            rsum += __shfl_xor(rsum, 1);
            rsum += __shfl_xor(rsum, 2);
            rsum += __shfl_xor(rsum, 4);
            rsum += __shfl_xor(rsum, 8);
            lrow[j] = lrow[j] * alpha + rsum;
            mrow[j] = mnew;
            o0[j] *= alpha;
            o1[j] *= alpha;
            p[j] = pj;
        }
        // ---- P through LDS: C/D layout -> A-fragment layout ----
        __syncthreads();
#pragma unroll
        for (int j = 0; j < 8; ++j)
            lp[(j + rowOff) * 17 + l16] = p[j];
        __syncthreads();
        // ---- O += P * V ----
        const float* vb = v + ((size_t)bh * TT + s0) * HSZ;
#pragma unroll
        for (int kk = 0; kk < 4; ++kk) {
            v2f pa = load_a_frag(lp, 17, kk * 4, l16, hiOff);
            v2f b0 = load_b_frag(vb, HSZ, kk * 4, 0,  l16, hiOff);
            v2f b1 = load_b_frag(vb, HSZ, kk * 4, 16, l16, hiOff);
            o0 = wmma_f32(pa, b0, o0);
            o1 = wmma_f32(pa, b1, o1);
        }
        __syncthreads();
    }
    // ---- normalize and scatter into concat-head layout [B,T,C] ----
#pragma unroll
    for (int j = 0; j < 8; ++j) {
        const float inv = 1.0f / lrow[j];
        const int   row = t0 + j + rowOff;
        float* orow = o + ((size_t)b * TT + row) * CC + head * HSZ;
        orow[l16]      = o0[j] * inv;
        orow[16 + l16] = o1[j] * inv;
    }
}

// ---------------------------------------------------------------------------
// Generic GEMM: out[m,n] = (relu?)(A[m,:]*W[:,n] + bias[n]) (+ res[m,n])
// One wave computes a 32x64 tile: two 16-row sub-tiles share every B
// fragment (8 accumulators, ~10.6 FLOP/byte vs 6.4 at 16 rows).
// Weight slab prefetched 64 K-elements ahead (global_prefetch_b8).
// ---------------------------------------------------------------------------
__global__ void gemm_kernel(const float* __restrict__ A, const float* __restrict__ W,
                            const float* __restrict__ bias, const float* __restrict__ res,
                            float* __restrict__ out, int K, int N, int relu) {
    const int lane   = threadIdx.x;
    const int l16    = lane & 15;
    const int hiOff  = (lane >> 4) ? 2 : 0;
    const int rowOff = (lane >> 4) ? 8 : 0;
    const int m0     = blockIdx.x * 32;
    const int n0     = blockIdx.y * 64;

    const float* A0 = A + (size_t)m0 * K;
    const float* A1 = A0 + (size_t)16 * K;
    v8f acc0[4] = {};
    v8f acc1[4] = {};
    for (int k0 = 0; k0 < K; k0 += 4) {
        v2f a0 = load_a_frag(A0, K, k0, l16, hiOff);
        v2f a1 = load_a_frag(A1, K, k0, l16, hiOff);
        if (k0 + 64 < K)   // stream weights ahead of use
            __builtin_prefetch(W + (size_t)(k0 + 64 + hiOff) * N + n0 + l16, 0, 0);
#pragma unroll
        for (int i = 0; i < 4; ++i) {
            v2f bf = load_b_frag(W, N, k0, n0 + i * 16, l16, hiOff);
            acc0[i] = wmma_f32(a0, bf, acc0[i]);
            acc1[i] = wmma_f32(a1, bf, acc1[i]);
        }
    }
#pragma unroll
    for (int i = 0; i < 4; ++i) {
        const float bb = bias[n0 + i * 16 + l16];
#pragma unroll
        for (int s = 0; s < 2; ++s) {
            const v8f& acc = s ? acc1[i] : acc0[i];
#pragma unroll
            for (int j = 0; j < 8; ++j) {
                const int m = m0 + s * 16 + j + rowOff;
                const size_t idx = (size_t)m * N + n0 + i * 16 + l16;
                float val = acc[j] + bb;
                if (relu) val = fmaxf(val, 0.0f);
                if (res)  val += res[idx];
                out[idx] = val;
            }
        }
    }
}

// ---------------------------------------------------------------------------
// Host launcher
// ---------------------------------------------------------------------------
extern "C" void kernel_launch(void* const* d_in, const int* in_sizes, int n_in,
                              void* d_out, int out_size, void* d_ws, size_t ws_size,
                              hipStream_t stream) {
    const float* x   = (const float*)d_in[0];
    const float* Wq  = (const float*)d_in[1];
    const float* bq  = (const float*)d_in[2];
    const float* Wk  = (const float*)d_in[3];
    const float* bk  = (const float*)d_in[4];
    const float* Wv  = (const float*)d_in[5];
    const float* bv  = (const float*)d_in[6];
    const float* Wo  = (const float*)d_in[7];
    const float* bo  = (const float*)d_in[8];
    const float* W1  = (const float*)d_in[9];
    const float* b1  = (const float*)d_in[10];
    const float* W2  = (const float*)d_in[11];
    const float* b2  = (const float*)d_in[12];
    const float* g1  = (const float*)d_in[13];
    const float* be1 = (const float*)d_in[14];
    const float* g2  = (const float*)d_in[15];
    const float* be2 = (const float*)d_in[16];
    float* out = (float*)d_out;

    const size_t BT  = (size_t)BB * TT;      // 4096 rows
    const size_t ROW = (size_t)BT * CC;      // 2M floats per activation buffer

    float* h  = (float*)d_ws;     // LN output (reused for LN2)
    float* q  = h  + ROW;
    float* k  = q  + ROW;
    float* v  = k  + ROW;
    float* o  = v  + ROW;
    float* x2 = o  + ROW;
    float* f1 = q;                // FFN hidden (8M floats) reuses dead q..o

    // 1) h = LN(x)
    ln_kernel<<<dim3((unsigned)BT), dim3(32), 0, stream>>>(x, g1, be1, h);
    // 2) q,k,v = h @ W{q,k,v} + bias   (per head)
    qkv_kernel<<<dim3((unsigned)(BT / 32), HH, 3), dim3(32), 0, stream>>>(
        h, Wq, Wk, Wv, bq, bk, bv, q, k, v);
    // 3) o = causal_softmax(q k^T) v   (flash-style, no score materialization)
    attn_kernel<<<dim3(TT / 16, BB * HH), dim3(32), 0, stream>>>(q, k, v, o);
    // 4) x2 = x + o @ Wo + bo
    gemm_kernel<<<dim3((unsigned)(BT / 32), CC / 64), dim3(32), 0, stream>>>(
        o, Wo, bo, x, x2, CC, CC, 0);
    // 5) h = LN(x2)
    ln_kernel<<<dim3((unsigned)BT), dim3(32), 0, stream>>>(x2, g2, be2, h);
    // 6) f1 = relu(h @ W1 + b1)
    gemm_kernel<<<dim3((unsigned)(BT / 32), DFF / 64), dim3(32), 0, stream>>>(
        h, W1, b1, nullptr, f1, CC, DFF, 1);
    // 7) out = x2 + f1 @ W2 + b2
    gemm_kernel<<<dim3((unsigned)(BT / 32), CC / 64), dim3(32), 0, stream>>>(
        f1, W2, b2, x2, out, DFF, CC, 0);
}